// SpatialEmbLossTf_13408887898333
// MI455X (gfx1250) — compile-verified
//
#include <hip/hip_runtime.h>
#include <hip/hip_bf16.h>
#include <stdint.h>

#define K_CLS 32
#define CHUNK 8192u          // elements per LDS chunk (64 KB of the 320 KB WGP LDS)
#define BTHREADS 256

typedef float v2f __attribute__((ext_vector_type(2)));
typedef float v8f __attribute__((ext_vector_type(8)));

struct Stats {
    int   counts[K_CLS];
    float centerSum[K_CLS * 3];
    float sigmaSum, varSum, seedSum, instSum;
    float s_mean, s_val;
    float Ek[K_CLS];
    float invEk[K_CLS];
    int   offs[K_CLS];
};

// ---------------------------------------------------------------------------
// Exact fp32 wave reduction through the matrix pipe:
// D = A(16x4) * B(4x16) + C with B == ones  =>  D[m][n] = sum_k A[m][k].
// Summing one column of D (8 VGPRs) + one xor-16 shuffle = 32-lane sum.
// Layout-agnostic because B is all ones.
// ---------------------------------------------------------------------------
__device__ __forceinline__ float wave_reduce_wmma(float x) {
    v2f a; a.x = x;    a.y = 0.0f;
    v2f b; b.x = 1.0f; b.y = 1.0f;
    v8f c = {};
    v8f d = __builtin_amdgcn_wmma_f32_16x16x4_f32(
        false, a, false, b, (short)0, c, false, false);
    float s = d[0] + d[1] + d[2] + d[3] + d[4] + d[5] + d[6] + d[7];
    s += __shfl_xor(s, 16, 32);
    return s;
}

// Descending-order key: ascending uint order of key <=> descending float.
__device__ __forceinline__ uint32_t descKey(float f) {
    uint32_t u = __float_as_uint(f);
    uint32_t m = (u >> 31) ? ~u : (u | 0x80000000u);
    return ~m;
}

__device__ __forceinline__ unsigned lb64(const uint64_t* a, unsigned lo, unsigned hi, uint64_t v) {
    while (lo < hi) { unsigned mid = (lo + hi) >> 1; if (a[mid] <  v) lo = mid + 1; else hi = mid; }
    return lo;
}
__device__ __forceinline__ unsigned ub64(const uint64_t* a, unsigned lo, unsigned hi, uint64_t v) {
    while (lo < hi) { unsigned mid = (lo + hi) >> 1; if (a[mid] <= v) lo = mid + 1; else hi = mid; }
    return lo;
}

// jaccard at inclusive sorted position j with cum positives; j==-1 -> 0.
__device__ __forceinline__ float jacc(float p, int j, int cum) {
    if (j < 0) return 0.0f;
    float u = p + (float)(j + 1 - cum);
    return 1.0f - (p - (float)cum) / u;
}

// ---------------------------------------------------------------------------
__global__ void init_k(Stats* st) {
    int nw = (int)((sizeof(Stats) + 3) / 4);
    for (int i = threadIdx.x; i < nw; i += blockDim.x) ((int*)st)[i] = 0;
}

// Pass A: A_i = sum_d(tanh(pred_d)+spatial_d), per-class segment sums, sigma sum.
__global__ __launch_bounds__(BTHREADS) void pass_a(const float* __restrict__ inp,
                                                   const float* __restrict__ pred,
                                                   const int*   __restrict__ cls,
                                                   float* __restrict__ A,
                                                   Stats* st, int n) {
    __shared__ int   sCnt[K_CLS];
    __shared__ float sCen[K_CLS * 3];
    __shared__ float sWave[8];
    int tid = threadIdx.x;
    if (tid < K_CLS)     sCnt[tid] = 0;
    if (tid < K_CLS * 3) sCen[tid] = 0.0f;
    __syncthreads();

    int i = blockIdx.x * blockDim.x + tid;
    float sig = 0.0f;
    if (i < n) {
        float x0 = inp[i * 9 + 1], x1 = inp[i * 9 + 2], x2 = inp[i * 9 + 7];
        float sp0 = (x0 + 4.0f) / 8.0f;
        float sp1 = (x1 + 4.0f) / 8.0f;
        float sp2 = (x2 + 600.0f) / 1200.0f;
        float p0 = pred[i * 5 + 0], p1 = pred[i * 5 + 1], p2 = pred[i * 5 + 2];
        A[i] = (tanhf(p0) + sp0) + (tanhf(p1) + sp1) + (tanhf(p2) + sp2);
        sig = pred[i * 5 + 3];
        int c = cls[i];
        atomicAdd(&sCnt[c], 1);
        atomicAdd(&sCen[c * 3 + 0], sp0);
        atomicAdd(&sCen[c * 3 + 1], sp1);
        atomicAdd(&sCen[c * 3 + 2], sp2);
    }
    float wsum = wave_reduce_wmma(sig);      // all lanes converged: EXEC all ones
    int lane = tid & 31, wave = tid >> 5;
    if (lane == 0) sWave[wave] = wsum;
    __syncthreads();
    if (tid == 0) {
        float t = 0.0f;
        int nw = blockDim.x >> 5;
        for (int w = 0; w < nw; ++w) t += sWave[w];
        atomicAdd(&st->sigmaSum, t);
    }
    if (tid < K_CLS && sCnt[tid]) atomicAdd(&st->counts[tid], sCnt[tid]);
    if (tid < K_CLS * 3)          atomicAdd(&st->centerSum[tid], sCen[tid]);
}

__global__ void finalize_stats(Stats* st, int n) {
    int t = threadIdx.x;  // one wave
    float s_mean = st->sigmaSum / (float)n;
    float s = expf(10.0f * s_mean);
    if (t == 0) {
        st->s_mean = s_mean;
        st->s_val  = s;
        int acc = 0;
        for (int k = 0; k < K_CLS; ++k) { st->offs[k] = acc; acc += st->counts[k]; }
    }
    if (t < K_CLS) {
        float cnt = (float)st->counts[t];
        float C = (st->centerSum[t * 3 + 0] + st->centerSum[t * 3 + 1] +
                   st->centerSum[t * 3 + 2]) / cnt;
        float E = expf(s * C);
        st->Ek[t]    = E;
        st->invEk[t] = 1.0f / E;
    }
}

// Pass B: B_i = exp(-s*A_i), sort keys, var + seed terms.
__global__ __launch_bounds__(BTHREADS) void pass_b(const float* __restrict__ pred,
                                                   const int*   __restrict__ cls,
                                                   const float* __restrict__ A,
                                                   uint64_t* __restrict__ keyG,
                                                   uint64_t* __restrict__ keyC,
                                                   Stats* st, int n, int sortN) {
    int i = blockIdx.x * blockDim.x + threadIdx.x;
    float vvar = 0.0f, vseed = 0.0f;
    float s = st->s_val, sm = st->s_mean;
    if (i < n) {
        float B = expf(-s * A[i]);
        int c = cls[i];
        uint32_t dm = descKey(B);
        keyG[i] = ((uint64_t)dm << 32) | (uint32_t)c;
        keyC[i] = ((uint64_t)(uint32_t)c << 32) | dm;
        float dmu = pred[i * 5 + 3] - sm;
        vvar = dmu * dmu / (float)st->counts[c];
        float seed = 1.0f / (1.0f + expf(-pred[i * 5 + 4]));
        float down = B * st->Ek[c];
        float dd = seed - down;
        vseed = dd * dd;
    } else if (i < sortN) {
        keyG[i] = ~0ULL;
        keyC[i] = ~0ULL;
    }
    float rv = wave_reduce_wmma(vvar);
    float rs = wave_reduce_wmma(vseed);
    if ((threadIdx.x & 31) == 0) {
        atomicAdd(&st->varSum,  rv);
        atomicAdd(&st->seedSum, rs);
    }
}

// ------------------------- bitonic sort (LDS-fused) -------------------------
// Pair index t -> (idx, idx|j): insert a zero at bit log2(j).
__device__ __forceinline__ unsigned pairIdx(unsigned t, unsigned j) {
    return ((t & ~(j - 1u)) << 1) | (t & (j - 1u));
}

// Global compare-exchange step for j >= CHUNK (pairs span LDS chunks).
// Only real pairs launched: half threads per array.
__global__ __launch_bounds__(BTHREADS) void bitonic_global_step(uint64_t* __restrict__ a0,
                                                                uint64_t* __restrict__ a1,
                                                                unsigned j, unsigned k,
                                                                unsigned L) {
    unsigned tid = blockIdx.x * blockDim.x + threadIdx.x;
    unsigned half = L >> 1;
    uint64_t* arr = (tid < half) ? a0 : a1;
    unsigned t = tid & (half - 1u);
    unsigned idx = pairIdx(t, j);
    unsigned ixj = idx | j;
    uint64_t x = arr[idx], y = arr[ixj];
    bool up = ((idx & k) == 0u);
    if (up ? (x > y) : (x < y)) { arr[idx] = y; arr[ixj] = x; }
}

// Fully sort each CHUNK-sized chunk in LDS: all stages k=2..CHUNK (91 steps fused).
__global__ __launch_bounds__(BTHREADS) void bitonic_local_sort(uint64_t* __restrict__ a0,
                                                               uint64_t* __restrict__ a1,
                                                               unsigned L) {
    __shared__ uint64_t sh[CHUNK];
    unsigned nChunks = L / CHUNK;
    unsigned blk = blockIdx.x;
    uint64_t* arr = (blk < nChunks) ? a0 : a1;
    unsigned chunk = (blk < nChunks) ? blk : blk - nChunks;
    unsigned base = chunk * CHUNK;
    for (unsigned i = threadIdx.x; i < CHUNK; i += blockDim.x) sh[i] = arr[base + i];
    __syncthreads();
    for (unsigned k = 2; k <= CHUNK; k <<= 1) {
        for (unsigned j = k >> 1; j > 0; j >>= 1) {
            for (unsigned t = threadIdx.x; t < (CHUNK >> 1); t += blockDim.x) {
                unsigned idx = pairIdx(t, j);
                unsigned ixj = idx | j;
                uint64_t x = sh[idx], y = sh[ixj];
                bool up = (((base + idx) & k) == 0u);   // global direction bit
                if (up ? (x > y) : (x < y)) { sh[idx] = y; sh[ixj] = x; }
            }
            __syncthreads();
        }
    }
    for (unsigned i = threadIdx.x; i < CHUNK; i += blockDim.x) arr[base + i] = sh[i];
}

// For stage k > CHUNK: fused tail steps j = CHUNK/2 .. 1 in LDS.
// Direction is uniform per chunk here (idx < CHUNK <= k/2 => idx & k == 0).
__global__ __launch_bounds__(BTHREADS) void bitonic_local_merge(uint64_t* __restrict__ a0,
                                                                uint64_t* __restrict__ a1,
                                                                unsigned k, unsigned L) {
    __shared__ uint64_t sh[CHUNK];
    unsigned nChunks = L / CHUNK;
    unsigned blk = blockIdx.x;
    uint64_t* arr = (blk < nChunks) ? a0 : a1;
    unsigned chunk = (blk < nChunks) ? blk : blk - nChunks;
    unsigned base = chunk * CHUNK;
    bool up = ((base & k) == 0u);
    for (unsigned i = threadIdx.x; i < CHUNK; i += blockDim.x) sh[i] = arr[base + i];
    __syncthreads();
    for (unsigned j = CHUNK >> 1; j > 0; j >>= 1) {
        for (unsigned t = threadIdx.x; t < (CHUNK >> 1); t += blockDim.x) {
            unsigned idx = pairIdx(t, j);
            unsigned ixj = idx | j;
            uint64_t x = sh[idx], y = sh[ixj];
            if (up ? (x > y) : (x < y)) { sh[idx] = y; sh[ixj] = x; }
        }
        __syncthreads();
    }
    for (unsigned i = threadIdx.x; i < CHUNK; i += blockDim.x) arr[base + i] = sh[i];
}

// ------------------------- Lovasz (all classes, one sort) -------------------
__global__ __launch_bounds__(BTHREADS) void lovasz_k(const uint64_t* __restrict__ keyG,
                                                     const uint64_t* __restrict__ keyC,
                                                     Stats* st, int n) {
    __shared__ float sE[K_CLS], sInv[K_CLS];
    __shared__ int   sOff[K_CLS], sCnt[K_CLS];
    int tid = threadIdx.x;
    if (tid < K_CLS) {
        sE[tid]   = st->Ek[tid];
        sInv[tid] = st->invEk[tid];
        sOff[tid] = st->offs[tid];
        sCnt[tid] = st->counts[tid];
    }
    __syncthreads();

    int g = blockIdx.x * blockDim.x + tid;
    float contrib = 0.0f;
    if (g < n) {
        uint64_t kg = keyG[g];
        uint32_t dm   = (uint32_t)(kg >> 32);
        uint32_t mono = ~dm;
        uint32_t u    = (mono & 0x80000000u) ? (mono & 0x7FFFFFFFu) : ~mono;
        float B = __uint_as_float(u);
        int   c = (int)(kg & 0xFFFFFFFFu);

        { // positive contribution for own class c
            int p = sCnt[c]; float fp = (float)p;
            unsigned o = (unsigned)sOff[c];
            uint64_t base = (uint64_t)(uint32_t)c << 32;
            int descRank = (int)(lb64(keyC, o, o + (unsigned)p, base | dm) - o);
            int m = p - 1 - descRank;                  // ascending-B rank among positives
            float t = sInv[c] - B;                     // neg ahead iff B_n > 1/E - B
            uint32_t dt = descKey(t);
            int nall = (int)lb64(keyG, 0u, (unsigned)n, (uint64_t)dt << 32);
            int ncls = (int)(lb64(keyC, o, o + (unsigned)p, base | dt) - o);
            int nneg = nall - ncls;
            int j = m + nneg, cum = m + 1;
            float e = 2.0f - 2.0f * B * sE[c];
            float rel = e > 0.0f ? e : 0.0f;
            contrib += rel * (jacc(fp, j, cum) - jacc(fp, j - 1, cum - 1));
        }
        for (int k = 0; k < K_CLS; ++k) {              // negative for other classes
            if (k == c) continue;
            int p = sCnt[k]; float fp = (float)p;
            unsigned o = (unsigned)sOff[k];
            uint64_t base = (uint64_t)(uint32_t)k << 32;
            int before_k = (int)(lb64(keyC, o, o + (unsigned)p, base | dm) - o);
            int q = g - before_k;                      // rank among negatives of class k
            float t = sInv[k] - B;                     // pos ahead iff B_p < 1/E - B
            uint32_t dt = descKey(t);
            int geCnt = (int)(ub64(keyC, o, o + (unsigned)p, base | dt) - o);
            int mq = p - geCnt;
            int j = q + mq, cum = mq;
            float e = 2.0f * B * sE[k];
            contrib += e * (jacc(fp, j, cum) - jacc(fp, j - 1, cum));
        }
    }
    float r = wave_reduce_wmma(contrib);
    if ((tid & 31) == 0) atomicAdd(&st->instSum, r);
}

__global__ void final_k(const Stats* __restrict__ st, float* __restrict__ out, int n) {
    if (threadIdx.x == 0 && blockIdx.x == 0) {
        out[0] = 1.0f  * (st->instSum / (float)K_CLS)
               + 10.0f * (st->varSum  / (float)K_CLS)
               + 1.0f  * (st->seedSum / (float)n);
    }
}

// ---------------------------------------------------------------------------
extern "C" void kernel_launch(void* const* d_in, const int* in_sizes, int n_in,
                              void* d_out, int out_size, void* d_ws, size_t ws_size,
                              hipStream_t stream) {
    const float* inp  = (const float*)d_in[1];   // input   (N,9)
    const float* pred = (const float*)d_in[2];   // pred    (N,5)
    const int*   cls  = (const int*)  d_in[4];   // classes (N,)
    int n = in_sizes[4];

    unsigned sortN = 1;
    while (sortN < (unsigned)n) sortN <<= 1;
    if (sortN < 2) sortN = 2;

    char* ws = (char*)d_ws;
    Stats* st = (Stats*)ws;
    float* A  = (float*)(ws + 4096);
    size_t offKeys = 4096 + (((size_t)n * 4 + 255) & ~(size_t)255);
    uint64_t* keyG = (uint64_t*)(ws + offKeys);
    uint64_t* keyC = (uint64_t*)(ws + offKeys + (size_t)sortN * 8);
    // ws footprint: 4KB + 4N + 16*sortN  (~5.3 MB at N=262144)

    const int blk = BTHREADS;
    init_k<<<1, 256, 0, stream>>>(st);
    pass_a<<<(n + blk - 1) / blk, blk, 0, stream>>>(inp, pred, cls, A, st, n);
    finalize_stats<<<1, 32, 0, stream>>>(st, n);
    pass_b<<<((int)sortN + blk - 1) / blk, blk, 0, stream>>>(pred, cls, A, keyG, keyC,
                                                             st, n, (int)sortN);

    if (sortN >= CHUNK) {
        // LDS-fused bitonic: 171 global passes -> 21 launches at sortN = 2^18.
        unsigned nChunks = sortN / CHUNK;
        unsigned mergeBlocks = 2u * nChunks;
        bitonic_local_sort<<<mergeBlocks, blk, 0, stream>>>(keyG, keyC, sortN);
        for (unsigned k = CHUNK << 1; k <= sortN && k != 0; k <<= 1) {
            for (unsigned j = k >> 1; j >= CHUNK; j >>= 1)
                bitonic_global_step<<<(sortN + blk - 1) / blk, blk, 0, stream>>>(
                    keyG, keyC, j, k, sortN);
            bitonic_local_merge<<<mergeBlocks, blk, 0, stream>>>(keyG, keyC, k, sortN);
        }
    } else {
        for (unsigned k = 2; k <= sortN; k <<= 1)
            for (unsigned j = k >> 1; j > 0; j >>= 1)
                bitonic_global_step<<<(sortN + blk - 1) / blk, blk, 0, stream>>>(
                    keyG, keyC, j, k, sortN);
    }

    lovasz_k<<<(n + blk - 1) / blk, blk, 0, stream>>>(keyG, keyC, st, n);
    final_k<<<1, 1, 0, stream>>>(st, (float*)d_out, n);
}